// _SAGEConv_36687610642889
// MI455X (gfx1250) — compile-verified
//
#include <hip/hip_runtime.h>

#define N_NODES   100000
#define N_EDGES   1600000
#define D         128
#define N_CLASSES 10
#define N_GRAPHS  64

typedef __attribute__((ext_vector_type(2))) float v2f;
typedef __attribute__((ext_vector_type(8))) float v8f;

#define LDS_STRIDE 260  // 256 K-values + 4-float pad -> conflict-free b64 A reads

// ---------------------------------------------------------------------------
// Edge scatter: one wave32 per edge; each lane moves 4 channels (float4) and
// atomically accumulates into msum[dst]. Lane 0 bumps degree on layer 1.
// ---------------------------------------------------------------------------
__global__ __launch_bounds__(256) void sage_agg_kernel(
    const long long* __restrict__ src, const long long* __restrict__ dst,
    const float* __restrict__ feat, float* __restrict__ msum,
    float* __restrict__ deg, int do_deg)
{
    unsigned gid  = blockIdx.x * 256u + threadIdx.x;
    unsigned edge = gid >> 5;
    unsigned lane = gid & 31;
    if (edge >= N_EDGES) return;
    long long s = src[edge];
    long long d = dst[edge];
    float4 v = ((const float4*)(feat + (size_t)s * D))[lane];
    float* o = msum + (size_t)d * D + lane * 4;
    atomicAdd(o + 0, v.x);
    atomicAdd(o + 1, v.y);
    atomicAdd(o + 2, v.z);
    atomicAdd(o + 3, v.w);
    if (do_deg && lane == 0) atomicAdd(deg + d, 1.0f);
}

// ---------------------------------------------------------------------------
// Weight repack into WMMA B-fragment order.
// Wcat (256x128) = [Wl ; Wr].  Wp[(t*128 + n)*2 + h] = { Wcat[4t+2h][n],
// Wcat[4t+2h+1][n] }  (t = k0/4, h = khalf/2). Each lane's B fragment is then
// ONE contiguous global_load_b64; a wave's k-step load is 256 B contiguous.
// ---------------------------------------------------------------------------
__global__ __launch_bounds__(256) void pack_w_kernel(
    const float* __restrict__ Wl, const float* __restrict__ Wr,
    float2* __restrict__ Wp)
{
    int i = blockIdx.x * 256 + threadIdx.x;     // over 64*128*2 = 16384
    if (i >= 64 * D * 2) return;
    int h = i & 1;
    int n = (i >> 1) & (D - 1);
    int t = i >> 8;
    int k = t * 4 + h * 2;                      // k, k+1 stay in same matrix half
    const float* W0 = (k < D) ? (Wl + (size_t)k * D) : (Wr + (size_t)(k - D) * D);
    float2 v;
    v.x = W0[n];
    v.y = W0[D + n];
    Wp[i] = v;
}

// ---------------------------------------------------------------------------
// Fused SAGE layer: out = relu([msum*rdeg | root] @ Wcat + bias)
// Block = 256 threads = 8 waves; block computes 32 rows x 128 cols.
// Each wave: one 16-col tile, TWO 16-row tiles -> each B fragment feeds 2 wmmas.
// K = 256 accumulated with V_WMMA_F32_16X16X4_F32 (full fp32 precision).
// ---------------------------------------------------------------------------
__global__ __launch_bounds__(256) void sage_gemm_kernel(
    const float* msum, const float* __restrict__ deg,
    const float* __restrict__ root, const float2* __restrict__ Wp,
    const float* __restrict__ bias, float* out)
{
    __shared__ float lds[32 * LDS_STRIDE];
    __shared__ float rdeg[32];
    const unsigned tid  = threadIdx.x;
    const unsigned row0 = blockIdx.x * 32u;     // 100000 % 32 == 0

    if (tid < 32) rdeg[tid] = 1.0f / fmaxf(deg[row0 + tid], 1.0f);
    __syncthreads();

    // Stage A = [mean | root] for 32 rows into LDS.
    for (unsigned i = tid; i < 32 * D; i += 256) {
        unsigned m = i >> 7;
        unsigned k = i & (D - 1);
        unsigned r = row0 + m;
        lds[m * LDS_STRIDE + k]     = msum[(size_t)r * D + k] * rdeg[m];
        lds[m * LDS_STRIDE + D + k] = root[(size_t)r * D + k];
    }
    __syncthreads();

    const unsigned lane  = tid & 31;
    const unsigned n0    = (tid >> 5) * 16;      // wave's N tile
    const unsigned mrow  = lane & 15;            // A: M = lane%16
    const unsigned khalf = (lane >> 4) * 2;      // A/B: K-subgroup 0 or 2
    const unsigned ncol  = n0 + (lane & 15);     // B/C/D column
    const float*  lrow0 = &lds[mrow * LDS_STRIDE];
    const float*  lrow1 = &lds[(16 + mrow) * LDS_STRIDE];
    const float2* wp    = Wp + (size_t)ncol * 2 + (khalf >> 1);

    v8f acc0 = {}, acc1 = {};
#pragma unroll 8
    for (unsigned k0 = 0; k0 < 2 * D; k0 += 4) { // K = 256, step 4
        v2f b  = *(const v2f*)&wp[(size_t)(k0 >> 2) * 256];
        v2f a0 = *(const v2f*)(lrow0 + k0 + khalf);
        v2f a1 = *(const v2f*)(lrow1 + k0 + khalf);
        acc0 = __builtin_amdgcn_wmma_f32_16x16x4_f32(
            false, a0, false, b, (short)0, acc0, false, false);
        acc1 = __builtin_amdgcn_wmma_f32_16x16x4_f32(
            false, a1, false, b, (short)0, acc1, false, false);
    }

    // C/D layout: VGPR v -> M = v + 8*(lane/16), N = lane%16 (+n0).
    const float    bv    = bias[ncol];
    const unsigned mbase = (lane >> 4) * 8;
#pragma unroll
    for (int v = 0; v < 8; v++) {
        out[(size_t)(row0 + mbase + v) * D + ncol]      = fmaxf(acc0[v] + bv, 0.0f);
        out[(size_t)(row0 + 16 + mbase + v) * D + ncol] = fmaxf(acc1[v] + bv, 0.0f);
    }
}

// ---------------------------------------------------------------------------
// Final linear (128 -> 10) fused with global mean pool accumulation.
// One wave32 per node; wave-reduce each class dot product.
// ---------------------------------------------------------------------------
__global__ __launch_bounds__(256) void final_pool_kernel(
    const float* __restrict__ h, const float* __restrict__ Wf,
    const float* __restrict__ bf, const long long* __restrict__ batch,
    float* __restrict__ pool_sum, float* __restrict__ pool_cnt)
{
    unsigned gid  = blockIdx.x * 256u + threadIdx.x;
    unsigned node = gid >> 5;
    unsigned lane = gid & 31;
    if (node >= N_NODES) return;
    float4 hv = ((const float4*)(h + (size_t)node * D))[lane];
    long long g = batch[node];
#pragma unroll
    for (int c = 0; c < N_CLASSES; c++) {
        float p = hv.x * Wf[(lane * 4 + 0) * N_CLASSES + c]
                + hv.y * Wf[(lane * 4 + 1) * N_CLASSES + c]
                + hv.z * Wf[(lane * 4 + 2) * N_CLASSES + c]
                + hv.w * Wf[(lane * 4 + 3) * N_CLASSES + c];
        for (int off = 16; off > 0; off >>= 1) p += __shfl_down(p, off, 32);
        if (lane == 0) atomicAdd(&pool_sum[g * N_CLASSES + c], p + bf[c]);
    }
    if (lane == 0) atomicAdd(&pool_cnt[g], 1.0f);
}

__global__ void pool_div_kernel(const float* __restrict__ sum,
                                const float* __restrict__ cnt,
                                float* __restrict__ out)
{
    int i = blockIdx.x * blockDim.x + threadIdx.x;
    if (i < N_GRAPHS * N_CLASSES)
        out[i] = sum[i] / fmaxf(cnt[i / N_CLASSES], 1.0f);
}

// ---------------------------------------------------------------------------
extern "C" void kernel_launch(void* const* d_in, const int* in_sizes, int n_in,
                              void* d_out, int out_size, void* d_ws, size_t ws_size,
                              hipStream_t stream)
{
    (void)in_sizes; (void)n_in; (void)out_size; (void)ws_size;

    const float*     x     = (const float*)d_in[0];
    const long long* ei    = (const long long*)d_in[1];
    const long long* batch = (const long long*)d_in[2];
    const float* Wl1 = (const float*)d_in[3];
    const float* bl1 = (const float*)d_in[4];
    const float* Wr1 = (const float*)d_in[5];
    const float* Wl2 = (const float*)d_in[6];
    const float* bl2 = (const float*)d_in[7];
    const float* Wr2 = (const float*)d_in[8];
    const float* Wf  = (const float*)d_in[9];
    const float* bf  = (const float*)d_in[10];
    const long long* src = ei;            // edge_index[0]
    const long long* dst = ei + N_EDGES;  // edge_index[1]

    char* ws = (char*)d_ws;
    size_t off = 0;
    float*  msum = (float*)(ws + off);  off += (size_t)N_NODES * D * 4;   // 51.2 MB
    float*  h1   = (float*)(ws + off);  off += (size_t)N_NODES * D * 4;   // 51.2 MB
    float*  deg  = (float*)(ws + off);  off += (((size_t)N_NODES * 4 + 255) & ~(size_t)255);
    float2* Wp1  = (float2*)(ws + off); off += (size_t)64 * D * 2 * 8;    // 128 KB
    float2* Wp2  = (float2*)(ws + off); off += (size_t)64 * D * 2 * 8;    // 128 KB
    float*  psum = (float*)(ws + off);  off += (size_t)N_GRAPHS * N_CLASSES * 4;
    float*  pcnt = (float*)(ws + off);  off += 256;

    const int aggBlocks  = (N_EDGES * 32 + 255) / 256;   // 200000
    const int gemmBlocks = N_NODES / 32;                 // 3125
    const int poolBlocks = (N_NODES * 32 + 255) / 256;   // 12500
    const int packBlocks = (64 * D * 2 + 255) / 256;     // 64

    // ---- Weight repack (independent of data path) ----
    pack_w_kernel<<<packBlocks, 256, 0, stream>>>(Wl1, Wr1, Wp1);
    pack_w_kernel<<<packBlocks, 256, 0, stream>>>(Wl2, Wr2, Wp2);

    // ---- Layer 1 ----
    hipMemsetAsync(msum, 0, (size_t)N_NODES * D * 4, stream);
    hipMemsetAsync(deg,  0, (size_t)N_NODES * 4, stream);
    sage_agg_kernel<<<aggBlocks, 256, 0, stream>>>(src, dst, x, msum, deg, 1);
    sage_gemm_kernel<<<gemmBlocks, 256, 0, stream>>>(msum, deg, x, Wp1, bl1, h1);

    // ---- Layer 2 (reuse msum; h2 written over msum — block reads own rows first) ----
    hipMemsetAsync(msum, 0, (size_t)N_NODES * D * 4, stream);
    sage_agg_kernel<<<aggBlocks, 256, 0, stream>>>(src, dst, h1, msum, deg, 0);
    sage_gemm_kernel<<<gemmBlocks, 256, 0, stream>>>(msum, deg, h1, Wp2, bl2, msum);

    // ---- Final linear + global mean pool ----
    hipMemsetAsync(psum, 0, (size_t)(N_GRAPHS * N_CLASSES) * 4 + 256, stream);
    final_pool_kernel<<<poolBlocks, 256, 0, stream>>>(msum, Wf, bf, batch, psum, pcnt);
    pool_div_kernel<<<(N_GRAPHS * N_CLASSES + 255) / 256, 256, 0, stream>>>(
        psum, pcnt, (float*)d_out);
}